// PSRoIPool_79302276153932
// MI455X (gfx1250) — compile-verified
//
#include <hip/hip_runtime.h>
#include <hip/hip_bf16.h>
#include <stdint.h>

// PS-ROI pooling for MI455X (gfx1250, wave32).
// Direct per-bin rectangle sums (beats SAT on bytes moved; 103MB features are
// resident in the 192MB L2 after first touch -> pure L2-gather workload).
// Wave-wide f32 reduction implemented with V_WMMA_F32_16X16X4_F32 (B = ones).
// ROI row staged to LDS via global_load_async_to_lds_b32 (ASYNCcnt path).

typedef __attribute__((ext_vector_type(2))) float v2f;
typedef __attribute__((ext_vector_type(8))) float v8f;

typedef __attribute__((address_space(1))) int* gptr_i32;
typedef __attribute__((address_space(3))) int* lptr_i32;

#define PSROI_N    4
#define PSROI_C    392
#define PSROI_H    128
#define PSROI_W    128
#define PSROI_P    7
#define PSROI_G    7
#define PSROI_COUT 8
#define PSROI_SCALE 0.0625f

#if __has_builtin(__builtin_amdgcn_global_load_async_to_lds_b32)
#define PSROI_HAVE_ASYNC 1
#endif

// Exact 32-lane sum using the matrix pipe:
// A(16x4): lanes0-15 -> A[m][0]=v[m], lanes16-31 -> A[m][2]=v[m+16], K=1,3 zero.
// B = all ones (layout-independent). D[m][n] = v[m] + v[m+16].
// Lane L<16 holds D[0..7][L] in c[0..7]; lane L>=16 holds D[8..15][L-16].
// compsum(low half) + compsum(high half) (via xor-16 shuffle) = full wave sum.
__device__ __forceinline__ float wave_sum32(float v) {
#if __has_builtin(__builtin_amdgcn_wmma_f32_16x16x4_f32)
    v2f a;
    a.x = v;
    a.y = 0.0f;
    v2f bones;
    bones.x = 1.0f;
    bones.y = 1.0f;
    v8f c = {};
    c = __builtin_amdgcn_wmma_f32_16x16x4_f32(
        /*neg_a=*/false, a, /*neg_b=*/false, bones,
        /*c_mod=*/(short)0, c, /*reuse_a=*/false, /*reuse_b=*/false);
    float s = ((c[0] + c[1]) + (c[2] + c[3])) + ((c[4] + c[5]) + (c[6] + c[7]));
    s += __shfl_xor(s, 16, 32);
    return s;
#else
    #pragma unroll
    for (int off = 16; off > 0; off >>= 1) v += __shfl_xor(v, off, 32);
    return v;
#endif
}

__global__ __launch_bounds__(256) void psroi_pool_kernel(
    const float* __restrict__ feat,   // [4,392,128,128]
    const float* __restrict__ rois,   // [512,5]
    float* __restrict__ out)          // [512,8,7,7]
{
    const int bid  = blockIdx.x;           // r * 7 + ph
    const int r    = bid / PSROI_P;
    const int ph   = bid - r * PSROI_P;
    const int lane = threadIdx.x & 31;
    const int co   = threadIdx.x >> 5;     // wave id = output channel 0..7

    __shared__ float s_roi[8];

    if (threadIdx.x < 5) {
#ifdef PSROI_HAVE_ASYNC
        const float* gp = rois + r * 5 + threadIdx.x;
        // generic->AS1 / AS3 casts via integer: low 32 bits of a generic LDS
        // pointer are the LDS offset (ISA 10.2), global flat address == AS1.
        __builtin_amdgcn_global_load_async_to_lds_b32(
            (gptr_i32)(uintptr_t)gp,
            (lptr_i32)(uint32_t)(uintptr_t)&s_roi[threadIdx.x],
            /*offset=*/0, /*cpol=*/0);
#else
        s_roi[threadIdx.x] = rois[r * 5 + threadIdx.x];
#endif
    }
#ifdef PSROI_HAVE_ASYNC
#if __has_builtin(__builtin_amdgcn_s_wait_asynccnt)
    __builtin_amdgcn_s_wait_asynccnt(0);
#else
    asm volatile("s_wait_asynccnt 0" ::: "memory");
#endif
#endif
    __syncthreads();

    const int   b   = (int)s_roi[0];
    const float rsw = rintf(s_roi[1]) * PSROI_SCALE;
    const float rsh = rintf(s_roi[2]) * PSROI_SCALE;
    const float rew = (rintf(s_roi[3]) + 1.0f) * PSROI_SCALE;
    const float reh = (rintf(s_roi[4]) + 1.0f) * PSROI_SCALE;
    const float roi_w = fmaxf(rew - rsw, 0.1f);
    const float roi_h = fmaxf(reh - rsh, 0.1f);
    const float bin_w = roi_w / (float)PSROI_P;
    const float bin_h = roi_h / (float)PSROI_P;

    const float fH = (float)PSROI_H, fW = (float)PSROI_W;
    const int h0 = (int)fminf(fmaxf(floorf((float)ph * bin_h + rsh), 0.0f), fH);
    const int h1 = (int)fminf(fmaxf(ceilf(((float)ph + 1.0f) * bin_h + rsh), 0.0f), fH);
    const int hh = h1 - h0;

    // base offset of (batch b); channel filled in per pw below
    const int batch_off = b * PSROI_C * PSROI_H * PSROI_W;

    #pragma unroll
    for (int pw = 0; pw < PSROI_P; ++pw) {
        const int w0 = (int)fminf(fmaxf(floorf((float)pw * bin_w + rsw), 0.0f), fW);
        const int w1 = (int)fminf(fmaxf(ceilf(((float)pw + 1.0f) * bin_w + rsw), 0.0f), fW);
        const int ww   = w1 - w0;
        const int area = hh * ww;

        const int c = (co * PSROI_G + ph) * PSROI_G + pw;  // position-sensitive channel
        const float* __restrict__ base =
            feat + batch_off + (c * PSROI_H + h0) * PSROI_W + w0;

        // Exact division by ww via round-up magic (Granlund-Montgomery):
        // m = floor(2^32/d)+1; mulhi(i,m) == i/d exactly since i < 2^9 and the
        // error term i*(m - 2^32/d)/2^32 < 2^9/2^32 << 1/d.
        const uint32_t den    = (ww > 0) ? (uint32_t)ww : 1u;
        const uint32_t wmagic = (uint32_t)(0x100000000ULL / den) + 1u;

        // flattened rectangle; lanes stride -> coalesced row segments.
        // Unrolled x2 so two independent global loads are in flight per lane.
        float acc = 0.0f;
        int i = lane;
        for (; i + 32 < area; i += 64) {
            const uint32_t ia = (uint32_t)i;
            const uint32_t ib = (uint32_t)(i + 32);
            const uint32_t ha = __umulhi(ia, wmagic);
            const uint32_t hb = __umulhi(ib, wmagic);
            const uint32_t wa = ia - ha * den;
            const uint32_t wb = ib - hb * den;
            const float xa = base[ha * PSROI_W + wa];
            const float xb = base[hb * PSROI_W + wb];
            acc += xa;
            acc += xb;
        }
        if (i < area) {
            const uint32_t ia = (uint32_t)i;
            const uint32_t ha = __umulhi(ia, wmagic);
            const uint32_t wa = ia - ha * den;
            acc += base[ha * PSROI_W + wa];
        }

        // uniform control flow here: EXEC all ones for WMMA
        const float s = wave_sum32(acc);

        if (lane == 0) {
            const float v = (area > 0) ? (s / (float)area) : 0.0f;
            out[((r * PSROI_COUT + co) * PSROI_P + ph) * PSROI_P + pw] = v;
        }
    }
}

extern "C" void kernel_launch(void* const* d_in, const int* in_sizes, int n_in,
                              void* d_out, int out_size, void* d_ws, size_t ws_size,
                              hipStream_t stream) {
    (void)n_in; (void)out_size; (void)d_ws; (void)ws_size;
    const float* feat = (const float*)d_in[0];
    const float* rois = (const float*)d_in[1];
    float* out = (float*)d_out;

    const int R = in_sizes[1] / 5;          // 512
    const int grid = R * PSROI_P;           // one block per (roi, ph)
    psroi_pool_kernel<<<grid, 256, 0, stream>>>(feat, rois, out);
}